// HistogramEncoder_10033043603497
// MI455X (gfx1250) — compile-verified
//
#include <hip/hip_runtime.h>
#include <hip/hip_bf16.h>

#define NUM_BINS 10
#define HIDDEN   128
#define RANGE_HI 10.0f
#define LN_EPS   1e-5f

#define W1T_STRIDE 14    // w1 transposed [128][14]: (14c+ka) even -> aligned b64; banks conflict-free
#define W2T_STRIDE 34    // w2 chunk transposed [128][34]: (34c+ka) even; banks conflict-free
#define ACT_STRIDE 132   // activations [16][132]: rows float4-aligned (132*4 % 16 == 0)
#define KCHUNK     32    // layer-2 K staged in 4 chunks of 32

typedef __attribute__((ext_vector_type(2))) float v2f;
typedef __attribute__((ext_vector_type(8))) float v8f;

// ---------------------------------------------------------------------------
// Kernel 1: streaming per-row histogram. This is the roofline term:
// 4096x16384x4B = 256 MiB @ 23.3 TB/s => ~11 us; everything else is noise.
// One 256-thread workgroup per row; per-thread privatized LDS counters
// [bin][tid] (bank = tid%64, conflict-free); float counters (exact to 2^24);
// b128 global loads.
// ---------------------------------------------------------------------------
__global__ __launch_bounds__(256) void hist_kernel(const float* __restrict__ x,
                                                   float* __restrict__ h, int S) {
  __shared__ float cnt[NUM_BINS][256];
  const int tid = threadIdx.x;
  const int row = blockIdx.x;
  #pragma unroll
  for (int b = 0; b < NUM_BINS; ++b) cnt[b][tid] = 0.0f;
  __syncthreads();

  const float4* xr = (const float4*)(x + (size_t)row * S);
  const int n4 = S >> 2;
  for (int i = tid; i < n4; i += 256) {
    float4 v = xr[i];
    float vv[4] = {v.x, v.y, v.z, v.w};
    #pragma unroll
    for (int j = 0; j < 4; ++j) {
      float val = vv[j];
      if (val >= 0.0f && val <= RANGE_HI) {
        int b = (int)val;                      // val >= 0 -> trunc == floor
        if (b > NUM_BINS - 1) b = NUM_BINS - 1;
        cnt[b][tid] += 1.0f;
      }
    }
  }
  __syncthreads();

  for (int s = 128; s >= 1; s >>= 1) {
    if (tid < s) {
      #pragma unroll
      for (int b = 0; b < NUM_BINS; ++b) cnt[b][tid] += cnt[b][tid + s];
    }
    __syncthreads();
  }
  if (tid < NUM_BINS) {
    float tot = 0.0f;
    #pragma unroll
    for (int b = 0; b < NUM_BINS; ++b) tot += cnt[b][0];
    h[row * NUM_BINS + tid] = cnt[tid][0] / (tot + 1e-8f);
  }
}

// ---------------------------------------------------------------------------
// Kernel 2: fused 2-layer MLP with V_WMMA_F32_16X16X4_F32 (full f32
// precision; compute here is ~0.15 GFLOP, <<1% of runtime). One wave per
// 16-row batch tile; all WMMA code convergent (EXEC all ones).
//
// Weights are staged TRANSPOSED in LDS so every A- and B-fragment is a
// single aligned ds_load_b64 into an even VGPR pair (no gather movs).
//
// A-frag (16x4 f32): lane<16  -> M=lane,    v0=K(k0),   v1=K(k0+1)
//                    lane>=16 -> M=lane-16, v0=K(k0+2), v1=K(k0+3)
// B-frag (4x16) mirrors with lane%16 = N.
// D (16x16 f32): VGPR i -> row i + 8*(lane>=16), col = lane%16.
// ---------------------------------------------------------------------------
__global__ __launch_bounds__(32) void mlp_kernel(
    const float* __restrict__ h,
    const float* __restrict__ w1, const float* __restrict__ b1,
    const float* __restrict__ g1, const float* __restrict__ be1,
    const float* __restrict__ w2, const float* __restrict__ b2,
    const float* __restrict__ g2, const float* __restrict__ be2,
    float* __restrict__ out, int B) {
  __shared__ float hT[16][12];                 // histogram tile, K padded 10 -> 12
  __shared__ float w1t[HIDDEN][W1T_STRIDE];    // w1 transposed [col][K], K zero-padded
  __shared__ float w2t[HIDDEN][W2T_STRIDE];    // w2 chunk transposed [col][k-in-chunk]
  __shared__ float act[16][ACT_STRIDE];        // layer-1 activations
  __shared__ float act2[16][ACT_STRIDE];       // layer-2 pre-LN activations

  const int lane  = threadIdx.x;               // 0..31, one wave
  const int col16 = lane & 15;
  const int half  = lane >> 4;
  const int r0    = blockIdx.x * 16;

  // ---- stage w1 transposed into LDS (uniform trip counts) ----
  {
    const float4* w1v = (const float4*)w1;
    for (int i = lane; i < (NUM_BINS * HIDDEN) / 4; i += 32) {   // 320 = 10*32
      float4 v = w1v[i];
      int k = (i * 4) / HIDDEN, c = (i * 4) % HIDDEN;
      w1t[c + 0][k] = v.x; w1t[c + 1][k] = v.y;
      w1t[c + 2][k] = v.z; w1t[c + 3][k] = v.w;
    }
    for (int i = lane; i < 2 * HIDDEN; i += 32) {                // zero K=10,11
      w1t[i & (HIDDEN - 1)][NUM_BINS + (i >> 7)] = 0.0f;
    }
  }
  // ---- stage histogram tile (192 = 6*32, uniform) ----
  for (int i = lane; i < 16 * 12; i += 32) {
    int r = i / 12, k = i % 12;
    float v = 0.0f;
    if (k < NUM_BINS && (r0 + r) < B) v = h[(r0 + r) * NUM_BINS + k];
    hT[r][k] = v;
  }
  __syncthreads();

  // ---------------- layer 1: [16x12] x [12x128] ----------------
  {
    v8f acc[8] = {};
    #pragma unroll
    for (int k0 = 0; k0 < 12; k0 += 4) {
      const int ka = k0 + half * 2;
      v2f a = *(const v2f*)&hT[col16][ka];
      v2f bf[8];
      #pragma unroll
      for (int t = 0; t < 8; ++t)
        bf[t] = *(const v2f*)&w1t[t * 16 + col16][ka];   // one aligned b64 each
      #pragma unroll
      for (int t = 0; t < 8; ++t)
        acc[t] = __builtin_amdgcn_wmma_f32_16x16x4_f32(
            false, a, false, bf[t], (short)0, acc[t], false, false);
    }
    #pragma unroll
    for (int t = 0; t < 8; ++t) {
      const int col  = t * 16 + col16;
      const float bs = b1[col];
      #pragma unroll
      for (int i = 0; i < 8; ++i) {
        float v = acc[t][i] + bs;
        act[i + 8 * half][col] = v > 0.0f ? v : 0.0f;   // ReLU
      }
    }
  }
  __syncthreads();

  // ---------------- LayerNorm 1 (in place, float4) ----------------
  {
    const int r  = col16;
    const int c0 = half * 64;
    const float4* arow = (const float4*)&act[r][c0];
    float s = 0.0f, sq = 0.0f;
    #pragma unroll
    for (int c4 = 0; c4 < 16; ++c4) {
      float4 v = arow[c4];
      s  += v.x + v.y + v.z + v.w;
      sq += v.x * v.x + v.y * v.y + v.z * v.z + v.w * v.w;
    }
    s  += __shfl_xor(s, 16, 32);
    sq += __shfl_xor(sq, 16, 32);
    const float m   = s * (1.0f / HIDDEN);
    const float var = sq * (1.0f / HIDDEN) - m * m;
    const float inv = rsqrtf(var + LN_EPS);
    const float4* g4 = (const float4*)(g1 + c0);
    const float4* b4 = (const float4*)(be1 + c0);
    float4* awr = (float4*)&act[r][c0];
    #pragma unroll
    for (int c4 = 0; c4 < 16; ++c4) {
      float4 v = arow[c4], gg = g4[c4], bb = b4[c4];
      v.x = (v.x - m) * inv * gg.x + bb.x;
      v.y = (v.y - m) * inv * gg.y + bb.y;
      v.z = (v.z - m) * inv * gg.z + bb.z;
      v.w = (v.w - m) * inv * gg.w + bb.w;
      awr[c4] = v;
    }
  }
  __syncthreads();

  // ---------------- layer 2: [16x128] x [128x128], K chunked ----------------
  {
    v8f acc[8] = {};
    #pragma unroll 1
    for (int kc = 0; kc < HIDDEN; kc += KCHUNK) {
      // stage w2[kc..kc+32)[*] transposed: 1024 float4 = 32/lane, coalesced b128
      const float4* w2v = (const float4*)(w2 + (size_t)kc * HIDDEN);
      for (int i = lane; i < (KCHUNK * HIDDEN) / 4; i += 32) {
        float4 v = w2v[i];
        int k = (i * 4) / HIDDEN, c = (i * 4) % HIDDEN;
        w2t[c + 0][k] = v.x; w2t[c + 1][k] = v.y;
        w2t[c + 2][k] = v.z; w2t[c + 3][k] = v.w;
      }
      __syncthreads();
      #pragma unroll
      for (int k0 = 0; k0 < KCHUNK; k0 += 4) {
        const int ka = k0 + half * 2;
        v2f a = *(const v2f*)&act[col16][kc + ka];       // aligned b64
        v2f bf[8];
        #pragma unroll
        for (int t = 0; t < 8; ++t)
          bf[t] = *(const v2f*)&w2t[t * 16 + col16][ka]; // aligned b64, no conflicts
        #pragma unroll
        for (int t = 0; t < 8; ++t)
          acc[t] = __builtin_amdgcn_wmma_f32_16x16x4_f32(
              false, a, false, bf[t], (short)0, acc[t], false, false);
      }
      __syncthreads();   // before next chunk overwrites w2t
    }
    #pragma unroll
    for (int t = 0; t < 8; ++t) {
      const int col  = t * 16 + col16;
      const float bs = b2[col];
      #pragma unroll
      for (int i = 0; i < 8; ++i) {
        float v = acc[t][i] + bs;
        act2[i + 8 * half][col] = v > 0.0f ? v : 0.0f;  // ReLU
      }
    }
  }
  __syncthreads();

  // ---------------- LayerNorm 2 + store (float4) ----------------
  {
    const int r  = col16;
    const int c0 = half * 64;
    const float4* arow = (const float4*)&act2[r][c0];
    float s = 0.0f, sq = 0.0f;
    #pragma unroll
    for (int c4 = 0; c4 < 16; ++c4) {
      float4 v = arow[c4];
      s  += v.x + v.y + v.z + v.w;
      sq += v.x * v.x + v.y * v.y + v.z * v.z + v.w * v.w;
    }
    s  += __shfl_xor(s, 16, 32);
    sq += __shfl_xor(sq, 16, 32);
    const float m   = s * (1.0f / HIDDEN);
    const float var = sq * (1.0f / HIDDEN) - m * m;
    const float inv = rsqrtf(var + LN_EPS);
    if (r0 + r < B) {
      const float4* g4 = (const float4*)(g2 + c0);
      const float4* b4 = (const float4*)(be2 + c0);
      float4* orow = (float4*)(out + (size_t)(r0 + r) * HIDDEN + c0);
      for (int c4 = 0; c4 < 16; ++c4) {
        float4 v = arow[c4], gg = g4[c4], bb = b4[c4];
        v.x = (v.x - m) * inv * gg.x + bb.x;
        v.y = (v.y - m) * inv * gg.y + bb.y;
        v.z = (v.z - m) * inv * gg.z + bb.z;
        v.w = (v.w - m) * inv * gg.w + bb.w;
        orow[c4] = v;
      }
    }
  }
}

extern "C" void kernel_launch(void* const* d_in, const int* in_sizes, int n_in,
                              void* d_out, int out_size, void* d_ws, size_t ws_size,
                              hipStream_t stream) {
  const float* workloads = (const float*)d_in[0];
  const float* w1  = (const float*)d_in[1];
  const float* b1  = (const float*)d_in[2];
  const float* g1  = (const float*)d_in[3];
  const float* be1 = (const float*)d_in[4];
  const float* w2  = (const float*)d_in[5];
  const float* b2  = (const float*)d_in[6];
  const float* g2  = (const float*)d_in[7];
  const float* be2 = (const float*)d_in[8];
  float* out = (float*)d_out;

  const int B = out_size / HIDDEN;     // 4096
  const int S = in_sizes[0] / B;       // 16384
  float* h = (float*)d_ws;             // [B, NUM_BINS] intermediate

  hipLaunchKernelGGL(hist_kernel, dim3(B), dim3(256), 0, stream, workloads, h, S);
  const int nblk = (B + 15) / 16;
  hipLaunchKernelGGL(mlp_kernel, dim3(nblk), dim3(32), 0, stream,
                     h, w1, b1, g1, be1, w2, b2, g2, be2, out, B);
}